// GraphAttention_11355893530634
// MI455X (gfx1250) — compile-verified
//
#include <hip/hip_runtime.h>
#include <hip/hip_bf16.h>
#include <math.h>

// ---------------------------------------------------------------------------
// GraphAttention on MI455X (gfx1250):
//   scores = tanh(x@w1 + b1) @ w2 + b2    (WMMA f32 16x16x4, w1 in LDS)
//   attn   = softmax(scores) over ALL rows (two-stage online reduction)
//   out    = segment_sum(x * attn, batch)  (sorted batch -> register acc)
// ---------------------------------------------------------------------------

typedef float v2f __attribute__((ext_vector_type(2)));
typedef float v8f __attribute__((ext_vector_type(8)));

#define DIM_D 256
#define DIM_H 128
#define TILES_PER_BLOCK 8   // 8 waves * 16 rows = 128 rows per workgroup
#define SEG_CHUNK 512

// ---------------------------------------------------------------------------
// Kernel 1: scores[n] = tanh(x[n,:] @ w1 + b1) @ w2 + b2
// One wave -> one 16-row tile. w1 (256x128 f32 = 128KB) staged in LDS,
// interleaved so B operand (4x16 K-major subtile) is one 8B ds load per lane.
// ---------------------------------------------------------------------------
__global__ __launch_bounds__(256) void ga_scores_kernel(
    const float* __restrict__ x, const float* __restrict__ w1,
    const float* __restrict__ b1, const float* __restrict__ w2,
    const float* __restrict__ b2, float* __restrict__ scores, int N) {
  __shared__ float w1s[DIM_D * DIM_H];  // 131072 bytes (<320KB WGP LDS)

  const int tid = threadIdx.x;
  // Cooperative LDS fill with pair-interleave: element (d,h) ->
  // w1s[((d>>1)*H + h)*2 + (d&1)]  so that (w1[k][n], w1[k+1][n]) is a v2f.
  for (int i = tid * 4; i < DIM_D * DIM_H; i += 256 * 4) {
    float4 v = *(const float4*)(w1 + i);
#pragma unroll
    for (int j = 0; j < 4; ++j) {
      int d = (i + j) >> 7;          // /128
      int h = (i + j) & (DIM_H - 1); // %128
      w1s[(((d >> 1) * DIM_H) + h) * 2 + (d & 1)] = ((const float*)&v)[j];
    }
  }
  __syncthreads();

  const int lane = tid & 31;
  const int wave = tid >> 5;
  const int tile = blockIdx.x * TILES_PER_BLOCK + wave;
  if (tile * 16 >= N) return;

  const int mrow = tile * 16 + (lane & 15);  // A-matrix row for this lane
  const int klo  = (lane >> 4) * 2;          // K sub-offset (0 or 2)
  const int ncol = lane & 15;                // B/C column within 16-tile

  // C init with b1 broadcast: element (r,lane) has N = nt*16 + (lane&15).
  v8f c[8];
#pragma unroll
  for (int nt = 0; nt < 8; ++nt) {
    float b1n = b1[nt * 16 + ncol];
    c[nt] = (v8f){b1n, b1n, b1n, b1n, b1n, b1n, b1n, b1n};
  }

  const float* xrow = x + (size_t)mrow * DIM_D;
  for (int k0 = 0; k0 < DIM_D; k0 += 4) {
    // A 16x4 f32: VGPR0/1 = K={k0+klo, k0+klo+1}, rows striped over lanes 0..15.
    v2f a = *(const v2f*)(xrow + k0 + klo);
    const int krow = ((k0 + klo) >> 1) * DIM_H;
#pragma unroll
    for (int nt = 0; nt < 8; ++nt) {
      v2f b = *(const v2f*)(&w1s[(krow + nt * 16 + ncol) * 2]);
      c[nt] = __builtin_amdgcn_wmma_f32_16x16x4_f32(
          /*neg_a=*/false, a, /*neg_b=*/false, b,
          /*c_mod=*/(short)0, c[nt], /*reuse_a=*/false, /*reuse_b=*/false);
    }
  }

  // tanh + dot with w2, accumulated per (lane, r) -> partial row sums.
  float p[8];
#pragma unroll
  for (int r = 0; r < 8; ++r) p[r] = 0.0f;
#pragma unroll
  for (int nt = 0; nt < 8; ++nt) {
    float w2n = w2[nt * 16 + ncol];
#pragma unroll
    for (int r = 0; r < 8; ++r) p[r] += tanhf(c[nt][r]) * w2n;
  }
  // Reduce over the 16 lanes sharing each row (lanes 0-15 -> M=r, 16-31 -> M=r+8).
#pragma unroll
  for (int r = 0; r < 8; ++r) {
    p[r] += __shfl_xor(p[r], 1, 16);
    p[r] += __shfl_xor(p[r], 2, 16);
    p[r] += __shfl_xor(p[r], 4, 16);
    p[r] += __shfl_xor(p[r], 8, 16);
  }
  if ((lane & 15) == 0) {
    const float bb = b2[0];
    const int mbase = tile * 16 + (lane >> 4) * 8;
#pragma unroll
    for (int r = 0; r < 8; ++r) scores[mbase + r] = p[r] + bb;
  }
}

// ---------------------------------------------------------------------------
// Kernel 2a/2b: online-softmax reduction (max, sum-exp) over scores.
// ---------------------------------------------------------------------------
__device__ __forceinline__ void merge_ms(float& m1, float& s1, float m2, float s2) {
  float M = fmaxf(m1, m2);
  float a = (m1 == M) ? s1 : s1 * __expf(m1 - M);
  float b = (m2 == M) ? s2 : s2 * __expf(m2 - M);
  m1 = M;
  s1 = a + b;
}

__global__ __launch_bounds__(256) void ga_reduce1_kernel(
    const float* __restrict__ scores, int N, float* __restrict__ partials) {
  float m = -INFINITY, s = 0.0f;
  for (int i = blockIdx.x * 256 + threadIdx.x; i < N; i += gridDim.x * 256) {
    float v = scores[i];
    if (v > m) { s = s * __expf(m - v) + 1.0f; m = v; }
    else        s += __expf(v - m);
  }
  __shared__ float ms[256], ss[256];
  ms[threadIdx.x] = m; ss[threadIdx.x] = s;
  __syncthreads();
  for (int off = 128; off > 0; off >>= 1) {
    if (threadIdx.x < off) {
      float m1 = ms[threadIdx.x], s1 = ss[threadIdx.x];
      merge_ms(m1, s1, ms[threadIdx.x + off], ss[threadIdx.x + off]);
      ms[threadIdx.x] = m1; ss[threadIdx.x] = s1;
    }
    __syncthreads();
  }
  if (threadIdx.x == 0) {
    partials[blockIdx.x * 2 + 0] = ms[0];
    partials[blockIdx.x * 2 + 1] = ss[0];
  }
}

__global__ __launch_bounds__(256) void ga_reduce2_kernel(
    const float* __restrict__ partials, float* __restrict__ finals) {
  __shared__ float ms[256], ss[256];
  ms[threadIdx.x] = partials[threadIdx.x * 2 + 0];
  ss[threadIdx.x] = partials[threadIdx.x * 2 + 1];
  __syncthreads();
  for (int off = 128; off > 0; off >>= 1) {
    if (threadIdx.x < off) {
      float m1 = ms[threadIdx.x], s1 = ss[threadIdx.x];
      merge_ms(m1, s1, ms[threadIdx.x + off], ss[threadIdx.x + off]);
      ms[threadIdx.x] = m1; ss[threadIdx.x] = s1;
    }
    __syncthreads();
  }
  if (threadIdx.x == 0) { finals[0] = ms[0]; finals[1] = ss[0]; }
}

// ---------------------------------------------------------------------------
// Kernel 2c: normalized weights w[n] = exp(scores[n] - M) / S
// ---------------------------------------------------------------------------
__global__ __launch_bounds__(256) void ga_weights_kernel(
    const float* __restrict__ scores, const float* __restrict__ finals,
    float* __restrict__ wgt, int N) {
  int i = blockIdx.x * 256 + threadIdx.x;
  if (i < N) {
    float rs = 1.0f / finals[1];
    wgt[i] = __expf(scores[i] - finals[0]) * rs;
  }
}

// ---------------------------------------------------------------------------
// Kernel 3: out zero-init + chunked weighted segment sum (batch is sorted).
// ---------------------------------------------------------------------------
__global__ __launch_bounds__(256) void ga_zero_kernel(float* __restrict__ out, int n) {
  int i = blockIdx.x * 256 + threadIdx.x;
  if (i < n) out[i] = 0.0f;
}

__global__ __launch_bounds__(256) void ga_segsum_kernel(
    const float* __restrict__ x, const int* __restrict__ batch,
    const float* __restrict__ wgt, float* __restrict__ out, int N) {
  const int d = threadIdx.x;  // one column per thread (D == 256 == blockDim)
  int n0 = blockIdx.x * SEG_CHUNK;
  if (n0 >= N) return;
  int n1 = min(N, n0 + SEG_CHUNK);
  int cur = batch[n0];
  float acc = 0.0f;
  for (int n = n0; n < n1; ++n) {
    int b = batch[n];
    if (b != cur) {
      atomicAdd(&out[(size_t)cur * DIM_D + d], acc);
      acc = 0.0f;
      cur = b;
    }
    // stream x with a look-ahead prefetch hint (global_prefetch_b8)
    if (n + 8 < n1) __builtin_prefetch(x + (size_t)(n + 8) * DIM_D + d, 0, 1);
    acc = fmaf(x[(size_t)n * DIM_D + d], wgt[n], acc);
  }
  atomicAdd(&out[(size_t)cur * DIM_D + d], acc);
}

// ---------------------------------------------------------------------------
// Host-side launcher
// ---------------------------------------------------------------------------
extern "C" void kernel_launch(void* const* d_in, const int* in_sizes, int n_in,
                              void* d_out, int out_size, void* d_ws, size_t ws_size,
                              hipStream_t stream) {
  const float* x  = (const float*)d_in[0];
  const int* batch = (const int*)d_in[1];
  const float* w1 = (const float*)d_in[2];
  const float* b1 = (const float*)d_in[3];
  const float* w2 = (const float*)d_in[4];
  const float* b2 = (const float*)d_in[5];
  float* out = (float*)d_out;

  const int N = in_sizes[1];  // number of rows (batch has N elements)

  // Workspace layout (floats): scores[N] | weights[N] | partials[512] | finals[2]
  float* scores   = (float*)d_ws;
  float* wgt      = scores + N;
  float* partials = wgt + N;
  float* finals   = partials + 512;

  // 1) attention scores via WMMA f32
  int tiles = (N + 15) / 16;
  int grid1 = (tiles + TILES_PER_BLOCK - 1) / TILES_PER_BLOCK;
  ga_scores_kernel<<<grid1, 256, 0, stream>>>(x, w1, b1, w2, b2, scores, N);

  // 2) global softmax normalization (max + sum-exp), then weights
  ga_reduce1_kernel<<<256, 256, 0, stream>>>(scores, N, partials);
  ga_reduce2_kernel<<<1, 256, 0, stream>>>(partials, finals);
  ga_weights_kernel<<<(N + 255) / 256, 256, 0, stream>>>(scores, finals, wgt, N);

  // 3) weighted per-graph sum (sorted segments -> register accumulation)
  ga_zero_kernel<<<(out_size + 255) / 256, 256, 0, stream>>>(out, out_size);
  ga_segsum_kernel<<<(N + SEG_CHUNK - 1) / SEG_CHUNK, 256, 0, stream>>>(
      x, batch, wgt, out, N);
}